// AttnFeat_27419071217999
// MI455X (gfx1250) — compile-verified
//
#include <hip/hip_runtime.h>

// ---------------------------------------------------------------------------
// AttnFeat (edge-conv attention) for MI455X / gfx1250, wave32.
//   B=16, N=4096, k=16 neighbors, D_IN=16, FILTERS=32, GROUPS=4 (8 ch/group)
// Bandwidth-bound (~142.6 MB of output); the edge GEMM (16x16 x 16x32 per
// point) runs on V_WMMA_F32_16X16X4_F32 (exact f32), one point per wave.
// GroupNorm needs per-(b,group) stats over the whole cloud -> two passes.
// A-fragments are gathered with lane-dependent *addresses* (off folded into
// the pointer), not with dynamically indexed register arrays, so no cndmask
// select trees appear on the WMMA dependency chain.
// ---------------------------------------------------------------------------

#define B_   16
#define N_   4096
#define KNB  16
#define DIN  16
#define FLT  32

typedef __attribute__((ext_vector_type(2))) float v2f;
typedef __attribute__((ext_vector_type(8))) float v8f;

__device__ __forceinline__ float lrelu(float x) { return x >= 0.f ? x : 0.2f * x; }
__device__ __forceinline__ float xor32(float v, int m) { return __shfl_xor(v, m, 32); }

// Edge GEMM for one point: D[16 nbr][32 f] = (x[n]-x[nbr]) x W_edge.
// A (16x16, M=nbr, K=d) consumed in 4 K-steps of 4; two 16-wide N tiles.
// f32 WMMA layouts (ISA 7.12.2):
//   A 16x4 : lanes 0-15 hold {K=0,K=1}, lanes 16-31 hold {K=2,K=3}, M=lane&15
//   B 4x16 : VGPR0 = row K=0 (lo lanes)/K=2 (hi lanes), VGPR1 = K=1/K=3, N=lane&15
//   C/D    : VGPR v = row v (lo lanes) / row v+8 (hi lanes), N=lane&15
// off = 2 for hi half-wave, 0 for lo; folded into load addresses (8B aligned).
__device__ __forceinline__ void edge_wmma(const float* __restrict__ xc,
                                          const float* __restrict__ xn,
                                          const float* __restrict__ We,
                                          int m, int off, v8f acc[2]) {
  v2f af[4];
#pragma unroll
  for (int s = 0; s < 4; ++s) {
    const float2 c = *(const float2*)(xc + 4 * s + off);
    const float2 n = *(const float2*)(xn + 4 * s + off);
    v2f a; a.x = c.x - n.x; a.y = c.y - n.y;
    af[s] = a;
  }
#pragma unroll
  for (int t = 0; t < 2; ++t) {
    v8f c = {0.f, 0.f, 0.f, 0.f, 0.f, 0.f, 0.f, 0.f};
#pragma unroll
    for (int s = 0; s < 4; ++s) {
      const int r = 4 * s + off;
      const int fc = 16 * t + m;
      v2f bf; bf.x = We[r * FLT + fc]; bf.y = We[(r + 1) * FLT + fc];
      c = __builtin_amdgcn_wmma_f32_16x16x4_f32(false, af[s], false, bf,
                                                (short)0, c, false, false);
    }
    acc[t] = c;
  }
}

// ---------------------------------------------------------------------------
// ws layout (floats): [0..63]=gn1_sum[b][g], [64..127]=gn1_sq,
//                     [128..191]=gn2_sum,    [192..255]=gn2_sq
// ---------------------------------------------------------------------------
__global__ __launch_bounds__(256) void zero_ws_kernel(float* ws) {
  ws[threadIdx.x] = 0.f;
}

__global__ __launch_bounds__(256) void stats_kernel(
    const float* __restrict__ x, const int* __restrict__ nn_idx,
    const float* __restrict__ W1, const float* __restrict__ We,
    const float* __restrict__ be, float* __restrict__ ws) {
  const int lane = threadIdx.x & 31;
  const int wave = threadIdx.x >> 5;
  const int n = blockIdx.x * 8 + wave;
  const int b = blockIdx.y;
  const int m = lane & 15;
  const int off = (lane >> 3) & 2;  // 0 for lanes 0-15, 2 for lanes 16-31

  __shared__ float s_st[16];
  if (threadIdx.x < 16) s_st[threadIdx.x] = 0.f;
  __syncthreads();

  const size_t row = (size_t)(b * N_ + n);
  const float* xc = x + row * DIN;
  const int nb = nn_idx[row * KNB + m];
  const float* xn = x + (size_t)(b * N_ + nb) * DIN;

  // --- GN1 stats: relu(x.W1), one channel per lane (f = lane) ---
  float nf = 0.f;
#pragma unroll
  for (int d = 0; d < DIN; ++d) nf = fmaf(xc[d], W1[d * FLT + lane], nf);
  nf = fmaxf(nf, 0.f);
  float nfs = nf, nfq = nf * nf;
  nfs += xor32(nfs, 1); nfq += xor32(nfq, 1);
  nfs += xor32(nfs, 2); nfq += xor32(nfq, 2);
  nfs += xor32(nfs, 4); nfq += xor32(nfq, 4);
  if ((lane & 7) == 0) {
    int g = lane >> 3;
    atomicAdd(&s_st[0 + g], nfs);
    atomicAdd(&s_st[4 + g], nfq);
  }

  // --- GN2 stats: edge_pre.We + be via WMMA ---
  v8f acc[2];
  edge_wmma(xc, xn, We, m, off, acc);
  float ls[2], lq[2];
#pragma unroll
  for (int t = 0; t < 2; ++t) {
    float fb = be[16 * t + m];
    float s = 0.f, q = 0.f;
#pragma unroll
    for (int v = 0; v < 8; ++v) {
      float e = acc[t][v] + fb;
      s += e; q += e * e;
    }
    ls[t] = s; lq[t] = q;
  }
#pragma unroll
  for (int t = 0; t < 2; ++t) {
    ls[t] += xor32(ls[t], 1);  lq[t] += xor32(lq[t], 1);
    ls[t] += xor32(ls[t], 2);  lq[t] += xor32(lq[t], 2);
    ls[t] += xor32(ls[t], 4);  lq[t] += xor32(lq[t], 4);
    ls[t] += xor32(ls[t], 16); lq[t] += xor32(lq[t], 16);
  }
  if ((lane & 0x17) == 0) {  // lanes 0 and 8
#pragma unroll
    for (int t = 0; t < 2; ++t) {
      int g = 2 * t + (lane >> 3);  // tile0 -> groups 0/1, tile1 -> 2/3
      atomicAdd(&s_st[8 + g], ls[t]);
      atomicAdd(&s_st[12 + g], lq[t]);
    }
  }
  __syncthreads();
  if (threadIdx.x < 16) {
    int sec = threadIdx.x >> 2, g = threadIdx.x & 3;
    atomicAdd(&ws[sec * 64 + b * 4 + g], s_st[threadIdx.x]);
  }
}

__global__ __launch_bounds__(256) void attn_kernel(
    const float* __restrict__ x, const float* __restrict__ mask,
    const int* __restrict__ nn_idx, const float* __restrict__ W1,
    const float* __restrict__ g1, const float* __restrict__ b1,
    const float* __restrict__ We, const float* __restrict__ be,
    const float* __restrict__ g2, const float* __restrict__ b2,
    const float* __restrict__ Wsf, const float* __restrict__ bsf,
    const float* __restrict__ Wng, const float* __restrict__ bng,
    const float* __restrict__ ws, float* __restrict__ outp,
    float* __restrict__ edgep) {
  const int lane = threadIdx.x & 31;
  const int wave = threadIdx.x >> 5;
  const int n = blockIdx.x * 8 + wave;
  const int b = blockIdx.y;
  const int m = lane & 15;
  const bool hi = lane >= 16;
  const int off = (lane >> 3) & 2;

  const float inv1 = 1.f / (float)(N_ * 8);
  const float inv2 = 1.f / (float)(N_ * KNB * 8);
  const float bn_scale = rsqrtf(1.f + 1e-3f);

  // GN1 stats for this lane's channel group
  const int g1i = lane >> 3;
  float mu1 = ws[0 * 64 + b * 4 + g1i] * inv1;
  float rs1 = rsqrtf(ws[1 * 64 + b * 4 + g1i] * inv1 - mu1 * mu1 + 1e-3f);
  // GN2 stats per N-tile
  float mu2[2], rs2[2];
#pragma unroll
  for (int t = 0; t < 2; ++t) {
    int g = 2 * t + ((lane >> 3) & 1);
    mu2[t] = ws[128 + b * 4 + g] * inv2;
    rs2[t] = rsqrtf(ws[192 + b * 4 + g] * inv2 - mu2[t] * mu2[t] + 1e-3f);
  }

  const size_t row = (size_t)(b * N_ + n);
  const float* xc = x + row * DIN;
  const int nb = nn_idx[row * KNB + m];
  const float* xn = x + (size_t)(b * N_ + nb) * DIN;
  const float mb_lane = -10000.f * mask[b * N_ + nb];

  // --- new_feat -> GN1 -> self attention scalar (wave reduce) ---
  float nf = 0.f;
#pragma unroll
  for (int d = 0; d < DIN; ++d) nf = fmaf(xc[d], W1[d * FLT + lane], nf);
  nf = fmaxf(nf, 0.f);
  float nfn = (nf - mu1) * rs1 * g1[lane] + b1[lane];
  float sa = nfn * Wsf[lane];
  sa += xor32(sa, 1); sa += xor32(sa, 2); sa += xor32(sa, 4);
  sa += xor32(sa, 8); sa += xor32(sa, 16);
  sa = (sa + bsf[0]) * bn_scale;

  // --- edge GEMM (WMMA) + GN2 ---
  v8f acc[2];
  edge_wmma(xc, xn, We, m, off, acc);
  float en[2][8];
#pragma unroll
  for (int t = 0; t < 2; ++t) {
    int fc = 16 * t + m;
    float sc = rs2[t] * g2[fc];
    float sh = b2[fc] - mu2[t] * sc;
    float fb = be[fc];
#pragma unroll
    for (int v = 0; v < 8; ++v) en[t][v] = (acc[t][v] + fb) * sc + sh;
  }

  // --- neighbor attention: na[v] = sum_f en[row v(+8)][f] * Wng[f] ---
  float na[8];
#pragma unroll
  for (int v = 0; v < 8; ++v)
    na[v] = en[0][v] * Wng[m] + en[1][v] * Wng[16 + m];
#pragma unroll
  for (int v = 0; v < 8; ++v) {
    na[v] += xor32(na[v], 1);
    na[v] += xor32(na[v], 2);
    na[v] += xor32(na[v], 4);
    na[v] += xor32(na[v], 8);
  }
  // assemble all 16 neighbor dots per lane (swap halves)
  float nrow[16];
#pragma unroll
  for (int v = 0; v < 8; ++v) {
    float oth = xor32(na[v], 16);
    if (hi) { nrow[v] = oth;   nrow[8 + v] = na[v]; }
    else    { nrow[v] = na[v]; nrow[8 + v] = oth;   }
  }

  // --- logits, masked softmax over 16 neighbors (all in registers) ---
  float l[16];
#pragma unroll
  for (int j = 0; j < 16; ++j) {
    float mbj = __shfl(mb_lane, j, 32);  // row j's mask bias lives in lane j
    float lg = sa + (nrow[j] + bng[0]) * bn_scale;
    l[j] = lrelu(lg) + mbj;
  }
  float mx = l[0];
#pragma unroll
  for (int j = 1; j < 16; ++j) mx = fmaxf(mx, l[j]);
  float w[16], den = 0.f;
#pragma unroll
  for (int j = 0; j < 16; ++j) { w[j] = __expf(l[j] - mx); den += w[j]; }
  float rden = 1.f / den;
#pragma unroll
  for (int j = 0; j < 16; ++j) w[j] *= rden;

  // --- vals[f] = sum_k w[k] * edge[k][f], then outputs ---
#pragma unroll
  for (int t = 0; t < 2; ++t) {
    float pv = 0.f;
    if (hi) {
#pragma unroll
      for (int v = 0; v < 8; ++v) pv += w[8 + v] * en[t][v];
    } else {
#pragma unroll
      for (int v = 0; v < 8; ++v) pv += w[v] * en[t][v];
    }
    pv += xor32(pv, 16);
    if (!hi) outp[row * FLT + 16 * t + m] = lrelu(pv);
  }

  // --- write normalized edge_feat (B,N,k,32) ---
  const int mb = hi ? 8 : 0;
#pragma unroll
  for (int t = 0; t < 2; ++t)
#pragma unroll
    for (int v = 0; v < 8; ++v)
      edgep[(row * KNB + (v + mb)) * FLT + 16 * t + m] = en[t][v];
}

extern "C" void kernel_launch(void* const* d_in, const int* in_sizes, int n_in,
                              void* d_out, int out_size, void* d_ws, size_t ws_size,
                              hipStream_t stream) {
  const float* x    = (const float*)d_in[0];
  const float* mask = (const float*)d_in[1];
  const int*   nn   = (const int*)d_in[2];
  const float* W1   = (const float*)d_in[3];
  const float* g1   = (const float*)d_in[4];
  const float* b1   = (const float*)d_in[5];
  const float* We   = (const float*)d_in[6];
  const float* be   = (const float*)d_in[7];
  const float* g2   = (const float*)d_in[8];
  const float* b2   = (const float*)d_in[9];
  const float* Wsf  = (const float*)d_in[10];
  const float* bsf  = (const float*)d_in[11];
  const float* Wng  = (const float*)d_in[12];
  const float* bng  = (const float*)d_in[13];
  float* ws = (float*)d_ws;
  float* outp = (float*)d_out;
  float* edgep = outp + (size_t)B_ * N_ * FLT;

  zero_ws_kernel<<<1, 256, 0, stream>>>(ws);
  stats_kernel<<<dim3(N_ / 8, B_), 256, 0, stream>>>(x, nn, W1, We, be, ws);
  attn_kernel<<<dim3(N_ / 8, B_), 256, 0, stream>>>(
      x, mask, nn, W1, g1, b1, We, be, g2, b2, Wsf, bsf, Wng, bng,
      ws, outp, edgep);
}